// TransformerAttentionModule_34488587387547
// MI455X (gfx1250) — compile-verified
//
#include <hip/hip_runtime.h>
#include <hip/hip_bf16.h>

typedef __attribute__((ext_vector_type(2))) float v2f;
typedef __attribute__((ext_vector_type(8))) float v8f;

#define GN 50000
#define GE 800000
#define GDIM 256
#define GH 8
#define GD 32
#define NTILES (GN / 16)   // 3125, exact

// ---------------------------------------------------------------------------
// Float atomic max via integer punning (portable, maps to global_atomic ops)
// ---------------------------------------------------------------------------
__device__ __forceinline__ void atomicMaxFloat(float* addr, float val) {
  if (val >= 0.0f) {
    atomicMax((int*)addr, __float_as_int(val));
  } else {
    atomicMin((unsigned int*)addr, (unsigned int)__float_as_int(val));
  }
}

// ---------------------------------------------------------------------------
// Kernel 1: fused QKV projection.  out[n, j] = sum_d x[n,d] * W[j,d] + b[j]
// grid = (3125 node tiles, 3 projections), block = 256 threads (8 waves).
// Each wave computes two 16x16 output tiles with chained f32 WMMA (K=4).
// ---------------------------------------------------------------------------
__global__ __launch_bounds__(256, 2)
void qkv_gemm_kernel(const float* __restrict__ x,
                     const float* __restrict__ Wq, const float* __restrict__ bq,
                     const float* __restrict__ Wk, const float* __restrict__ bk,
                     const float* __restrict__ Wv, const float* __restrict__ bv,
                     float* __restrict__ qo, float* __restrict__ ko,
                     float* __restrict__ vo) {
  __shared__ float As[16 * GDIM];   // 16 KB x-tile

  const int base = blockIdx.x * 16;

  const float* W;
  const float* bias;
  float* out;
  if (blockIdx.y == 0)      { W = Wq; bias = bq; out = qo; }
  else if (blockIdx.y == 1) { W = Wk; bias = bk; out = ko; }
  else                      { W = Wv; bias = bv; out = vo; }

  // Stage 16x256 fp32 x-tile into LDS (contiguous rows -> one big copy)
  const float4* xsrc = (const float4*)(x + (size_t)base * GDIM);
  for (int i = threadIdx.x; i < 16 * GDIM / 4; i += 256) {
    ((float4*)As)[i] = xsrc[i];
  }
  __syncthreads();

  const int lane = threadIdx.x & 31;
  const int wid  = threadIdx.x >> 5;
  const int m    = lane & 15;           // A-matrix row within tile
  const int kk   = (lane >> 4) << 1;    // K sub-offset: 0 (lanes 0-15) / 2 (16-31)

  #pragma unroll
  for (int c = 0; c < 2; ++c) {
    const int ct  = wid * 2 + c;        // column tile 0..15
    const int col = ct * 16 + (lane & 15);
    const float* Wrow = W + (size_t)col * GDIM;   // B[k][n] = W[j=n][k]

    v8f acc = {};
    #pragma unroll 16
    for (int k0 = 0; k0 < GDIM; k0 += 4) {
      v2f a = *(const v2f*)&As[m * GDIM + k0 + kk];
      v2f b = *(const v2f*)&Wrow[k0 + kk];
      acc = __builtin_amdgcn_wmma_f32_16x16x4_f32(
          false, a, false, b, (short)0, acc, false, false);
    }

    const float bb = bias[col];
    #pragma unroll
    for (int r = 0; r < 8; ++r) {
      const int row = base + r + ((lane >> 4) << 3);  // D layout: M=r / M=r+8
      out[(size_t)row * GDIM + col] = acc[r] + bb;
    }
  }
}

// ---------------------------------------------------------------------------
// Kernel 2: init segment buffers
// ---------------------------------------------------------------------------
__global__ void init_kernel(float* __restrict__ accum,
                            float* __restrict__ smax,
                            float* __restrict__ ssum) {
  const int idx = blockIdx.x * blockDim.x + threadIdx.x;
  if (idx < GN * GDIM) accum[idx] = 0.0f;
  if (idx < GN * GH) {
    smax[idx] = -__builtin_huge_valf();
    ssum[idx] = 0.0f;
  }
}

// ---------------------------------------------------------------------------
// Kernel 3: per-(edge,head) attention score + segment max.
// q/k/v (154 MB) are L2-resident on MI455X (192 MB L2) -> fast random gathers.
// 8 consecutive threads cover one edge's 8 heads (1 KB contiguous) -> coalesced.
// ---------------------------------------------------------------------------
__global__ __launch_bounds__(256)
void edge_scores_kernel(const float* __restrict__ q, const float* __restrict__ k,
                        const int* __restrict__ src, const int* __restrict__ dst,
                        float* __restrict__ scores, float* __restrict__ smax) {
  const int t = blockIdx.x * blockDim.x + threadIdx.x;
  if (t >= GE * GH) return;
  const int e = t >> 3;
  const int h = t & 7;
  const int s = src[e];
  const int d = dst[e];

  const float4* qp = (const float4*)(q + (size_t)s * GDIM + h * GD);
  const float4* kp = (const float4*)(k + (size_t)d * GDIM + h * GD);
  float acc = 0.0f;
  #pragma unroll
  for (int i = 0; i < GD / 4; ++i) {
    float4 a = qp[i], b = kp[i];
    acc += a.x * b.x + a.y * b.y + a.z * b.z + a.w * b.w;
  }
  acc *= 0.17677669529663687f;   // 1/sqrt(32)

  scores[t] = acc;
  atomicMaxFloat(&smax[(size_t)d * GH + h], acc);
}

// ---------------------------------------------------------------------------
// Kernel 4: e = exp(score - smax[dst]); ssum[dst] += e;
//           accum[dst] += e * v[src]   (normalization deferred to kernel 5)
// ---------------------------------------------------------------------------
__global__ __launch_bounds__(256)
void edge_accum_kernel(const float* __restrict__ v,
                       const int* __restrict__ src, const int* __restrict__ dst,
                       const float* __restrict__ scores,
                       const float* __restrict__ smax,
                       float* __restrict__ ssum, float* __restrict__ accum) {
  const int t = blockIdx.x * blockDim.x + threadIdx.x;
  if (t >= GE * GH) return;
  const int e = t >> 3;
  const int h = t & 7;
  const int s = src[e];
  const int d = dst[e];

  const float m  = smax[(size_t)d * GH + h];
  const float ex = __expf(scores[t] - m);

  atomicAdd(&ssum[(size_t)d * GH + h], ex);

  const float* vp = v + (size_t)s * GDIM + h * GD;
  float* ap = accum + (size_t)d * GDIM + h * GD;
  #pragma unroll
  for (int i = 0; i < GD; ++i) {
    atomicAdd(&ap[i], ex * vp[i]);
  }
}

// ---------------------------------------------------------------------------
// Kernel 5: out = (accum / ssum) @ Wo^T + bo   (normalize fused into LDS load)
// ---------------------------------------------------------------------------
__global__ __launch_bounds__(256, 2)
void out_gemm_kernel(const float* __restrict__ accum,
                     const float* __restrict__ ssum,
                     const float* __restrict__ Wo, const float* __restrict__ bo,
                     float* __restrict__ out) {
  __shared__ float As[16 * GDIM];

  const int base = blockIdx.x * 16;

  // Load + normalize: feature f belongs to head f>>5
  for (int i = threadIdx.x; i < 16 * GDIM; i += 256) {
    const int r = i >> 8;          // 0..15 (GDIM==256)
    const int f = i & (GDIM - 1);
    const int node = base + r;
    const float sden = ssum[(size_t)node * GH + (f >> 5)];
    const float val  = accum[(size_t)node * GDIM + f];
    As[i] = (sden > 0.0f) ? (val / sden) : 0.0f;
  }
  __syncthreads();

  const int lane = threadIdx.x & 31;
  const int wid  = threadIdx.x >> 5;
  const int m    = lane & 15;
  const int kk   = (lane >> 4) << 1;

  #pragma unroll
  for (int c = 0; c < 2; ++c) {
    const int ct  = wid * 2 + c;
    const int col = ct * 16 + (lane & 15);
    const float* Wrow = Wo + (size_t)col * GDIM;

    v8f acc = {};
    #pragma unroll 16
    for (int k0 = 0; k0 < GDIM; k0 += 4) {
      v2f a = *(const v2f*)&As[m * GDIM + k0 + kk];
      v2f b = *(const v2f*)&Wrow[k0 + kk];
      acc = __builtin_amdgcn_wmma_f32_16x16x4_f32(
          false, a, false, b, (short)0, acc, false, false);
    }

    const float bb = bo[col];
    #pragma unroll
    for (int r = 0; r < 8; ++r) {
      const int row = base + r + ((lane >> 4) << 3);
      out[(size_t)row * GDIM + col] = acc[r] + bb;
    }
  }
}

// ---------------------------------------------------------------------------
// Host launch
// ---------------------------------------------------------------------------
extern "C" void kernel_launch(void* const* d_in, const int* in_sizes, int n_in,
                              void* d_out, int out_size, void* d_ws, size_t ws_size,
                              hipStream_t stream) {
  (void)in_sizes; (void)n_in; (void)out_size; (void)ws_size;

  const float* x  = (const float*)d_in[0];
  const float* Wq = (const float*)d_in[1];
  const float* bq = (const float*)d_in[2];
  const float* Wk = (const float*)d_in[3];
  const float* bk = (const float*)d_in[4];
  const float* Wv = (const float*)d_in[5];
  const float* bv = (const float*)d_in[6];
  const float* Wo = (const float*)d_in[7];
  const float* bo = (const float*)d_in[8];
  const int* src  = (const int*)d_in[9];
  const int* dst  = (const int*)d_in[10];

  // Workspace layout (floats)
  float* ws = (float*)d_ws;
  const size_t NPROJ = (size_t)GN * GDIM;        // 12.8M
  float* q      = ws;
  float* k      = ws + NPROJ;
  float* v      = ws + 2 * NPROJ;
  float* accum  = ws + 3 * NPROJ;
  float* scores = ws + 4 * NPROJ;                // E*H = 6.4M
  float* smax   = scores + (size_t)GE * GH;      // N*H = 0.4M
  float* ssum   = smax + (size_t)GN * GH;

  // 1. QKV projections (WMMA)
  qkv_gemm_kernel<<<dim3(NTILES, 3), 256, 0, stream>>>(
      x, Wq, bq, Wk, bk, Wv, bv, q, k, v);

  // 2. Init segment buffers
  {
    const int total = GN * GDIM;
    init_kernel<<<(total + 255) / 256, 256, 0, stream>>>(accum, smax, ssum);
  }

  // 3. Scores + segment max
  {
    const int total = GE * GH;
    edge_scores_kernel<<<(total + 255) / 256, 256, 0, stream>>>(
        q, k, src, dst, scores, smax);
  }

  // 4. Softmax numerator accumulation
  {
    const int total = GE * GH;
    edge_accum_kernel<<<(total + 255) / 256, 256, 0, stream>>>(
        v, src, dst, scores, smax, ssum, accum);
  }

  // 5. Normalize + output projection (WMMA) -> d_out
  out_gemm_kernel<<<NTILES, 256, 0, stream>>>(
      accum, ssum, Wo, bo, (float*)d_out);
}